// HamiltonianDynamics_27977416966525
// MI455X (gfx1250) — compile-verified
//
#include <hip/hip_runtime.h>
#include <math.h>

typedef __attribute__((ext_vector_type(2))) float v2f;
typedef __attribute__((ext_vector_type(4))) float v4f;
typedef __attribute__((ext_vector_type(8))) float v8f;

#define NQF 524288.0f          // 8*256*256 elements per batch-half
#define INV_NQ (1.0f / 524288.0f)

// Padded LDS row strides (avoid 16-way bank conflicts on column-pattern reads)
#define S_S   8
#define W1_S  132
#define W1T_S 20
#define W2_S  132
#define W3_S  68
#define H12_S 132
#define H3_S  68
#define CW1_S 68
#define CW2_S 36
#define CW3_S 20
#define CL1_S 68
#define CL2_S 36

// Hardware tanh (TRANS32 op on CDNA5).
__device__ __forceinline__ float htanh(float x) {
#if __has_builtin(__builtin_amdgcn_tanhf)
  return __builtin_amdgcn_tanhf(x);
#else
  float r;
  asm volatile("v_tanh_f32 %0, %1\n\tv_nop\n\tv_nop" : "=v"(r) : "v"(x));
  return r;
#endif
}

// ---------------------------------------------------------------------------
// One 16x16 f32 WMMA tile: C += A(16xK) * B(Kx16), K % 4 == 0 (compile-time).
// Register layouts per CDNA5 ISA 7.12.2 (h = lane>>4, r = lane&15):
//   A 16x4 : vgpr v -> A[r][k0 + v + 2h]
//   B 4x16 : vgpr v -> B[k0 + v + 2h][r]
//   C 16x16: vgpr v -> C[v + 8h][r]
// ---------------------------------------------------------------------------
template <int K, typename FA, typename FB>
__device__ __forceinline__ v8f mm_tile(FA fa, FB fb, v8f acc) {
  const int lane = threadIdx.x & 31;
  const int h = lane >> 4;
  const int r = lane & 15;
#pragma unroll
  for (int k0 = 0; k0 < K; k0 += 4) {
    const int ka = k0 + 2 * h;
    v2f a, b;
    a[0] = fa(r, ka);
    a[1] = fa(r, ka + 1);
    b[0] = fb(ka, r);
    b[1] = fb(ka + 1, r);
    acc = __builtin_amdgcn_wmma_f32_16x16x4_f32(false, a, false, b,
                                                (short)0, acc, false, false);
  }
  return acc;
}

// ---------------------------------------------------------------------------
// Kernel 1: partial (sum, sumsq) over 16384-float chunks; 32 chunks/(b,half).
// Normal-temporal loads: populate L2 (192MB) so k3 re-reads hit cache.
// ---------------------------------------------------------------------------
__global__ __launch_bounds__(256)
void k1_reduce(const v4f* __restrict__ st4, float* __restrict__ pd) {
  const int t = threadIdx.x;
  const long long base4 = (long long)blockIdx.x * 4096;
  float s = 0.f, q = 0.f;
#pragma unroll
  for (int i = 0; i < 16; ++i) {
    v4f v = st4[base4 + t + 256 * i];
    s += v.x + v.y + v.z + v.w;
    q += v.x * v.x + v.y * v.y + v.z * v.z + v.w * v.w;
  }
  __shared__ float rs[256], rq[256];
  rs[t] = s; rq[t] = q;
  __syncthreads();
  for (int off = 128; off > 0; off >>= 1) {
    if (t < off) { rs[t] += rs[t + off]; rq[t] += rq[t + off]; }
    __syncthreads();
  }
  if (t == 0) {
    pd[blockIdx.x * 2 + 0] = rs[0];
    pd[blockIdx.x * 2 + 1] = rq[0];
  }
}

// ---------------------------------------------------------------------------
// gH: grad of hamiltonian MLP (2->128->128->64->1, tanh) wrt inputs.
// All operands live in (padded, pre-zeroed) LDS. 8 waves, WMMA tiles.
// ---------------------------------------------------------------------------
__device__ void run_gH(const float* S, float* H1, float* H2, float* H3,
                       float* G, const float* sW1, const float* sW1T,
                       const float* sW2, const float* sW3, const float* sW4,
                       const float* sB1, const float* sB2, const float* sB3) {
  const int t = threadIdx.x;
  const int wid = t >> 5;
  const int lane = t & 31;
  const int h = lane >> 4, r = lane & 15;

  // L1: H1 = tanh(S @ W1 + b1)  (K=2 zero-padded to 4) -> 16 tiles
  for (int tile = wid; tile < 16; tile += 8) {
    const int m0 = (tile & 1) * 16, n0 = (tile >> 1) * 16;
    v8f acc = {};
    acc = mm_tile<4>([&](int m, int k) { return S[(m0 + m) * S_S + k]; },
                     [&](int k, int n) { return sW1[k * W1_S + n0 + n]; }, acc);
#pragma unroll
    for (int v = 0; v < 8; ++v)
      H1[(m0 + v + 8 * h) * H12_S + n0 + r] = htanh(acc[v] + sB1[n0 + r]);
  }
  __syncthreads();

  // L2: H2 = tanh(H1 @ W2 + b2)  (K=128) -> 16 tiles
  for (int tile = wid; tile < 16; tile += 8) {
    const int m0 = (tile & 1) * 16, n0 = (tile >> 1) * 16;
    v8f acc = {};
    acc = mm_tile<128>([&](int m, int k) { return H1[(m0 + m) * H12_S + k]; },
                       [&](int k, int n) { return sW2[k * W2_S + n0 + n]; },
                       acc);
#pragma unroll
    for (int v = 0; v < 8; ++v)
      H2[(m0 + v + 8 * h) * H12_S + n0 + r] = htanh(acc[v] + sB2[n0 + r]);
  }
  __syncthreads();

  // L3: H3 = tanh(H2 @ W3 + b3)  (N=64, K=128) -> 8 tiles (1/wave)
  {
    const int m0 = (wid & 1) * 16, n0 = (wid >> 1) * 16;
    v8f acc = {};
    acc = mm_tile<128>([&](int m, int k) { return H2[(m0 + m) * H12_S + k]; },
                       [&](int k, int n) { return sW3[k * W3_S + n0 + n]; },
                       acc);
#pragma unroll
    for (int v = 0; v < 8; ++v)
      H3[(m0 + v + 8 * h) * H3_S + n0 + r] = htanh(acc[v] + sB3[n0 + r]);
  }
  __syncthreads();

  // Backward seed: d_z3 = W4 * (1 - H3^2), in place.
  for (int e = t; e < 32 * 64; e += 256) {
    const int m = e >> 6, n = e & 63;
    const float hv = H3[m * H3_S + n];
    H3[m * H3_S + n] = sW4[n] * (1.0f - hv * hv);
  }
  __syncthreads();

  // d_z2 = (d_z3 @ W3^T) * (1 - H2^2), in place.  (K=64) -> 16 tiles
  for (int tile = wid; tile < 16; tile += 8) {
    const int m0 = (tile & 1) * 16, n0 = (tile >> 1) * 16;
    v8f acc = {};
    acc = mm_tile<64>([&](int m, int k) { return H3[(m0 + m) * H3_S + k]; },
                      [&](int k, int n) { return sW3[(n0 + n) * W3_S + k]; },
                      acc);
#pragma unroll
    for (int v = 0; v < 8; ++v) {
      const int m = m0 + v + 8 * h, n = n0 + r;
      const float hv = H2[m * H12_S + n];
      H2[m * H12_S + n] = acc[v] * (1.0f - hv * hv);
    }
  }
  __syncthreads();

  // d_z1 = (d_z2 @ W2^T) * (1 - H1^2), in place.  (K=128) -> 16 tiles
  for (int tile = wid; tile < 16; tile += 8) {
    const int m0 = (tile & 1) * 16, n0 = (tile >> 1) * 16;
    v8f acc = {};
    acc = mm_tile<128>([&](int m, int k) { return H2[(m0 + m) * H12_S + k]; },
                       [&](int k, int n) { return sW2[(n0 + n) * W2_S + k]; },
                       acc);
#pragma unroll
    for (int v = 0; v < 8; ++v) {
      const int m = m0 + v + 8 * h, n = n0 + r;
      const float hv = H1[m * H12_S + n];
      H1[m * H12_S + n] = acc[v] * (1.0f - hv * hv);
    }
  }
  __syncthreads();

  // d_s = d_z1 @ W1^T  (N padded to 16 with zeros, K=128) -> 2 tiles
  if (wid < 2) {
    const int m0 = wid * 16;
    v8f acc = {};
    acc = mm_tile<128>([&](int m, int k) { return H1[(m0 + m) * H12_S + k]; },
                       [&](int k, int n) { return sW1T[k * W1T_S + n]; }, acc);
    if (r < 2) {
#pragma unroll
      for (int v = 0; v < 8; ++v) G[(m0 + v + 8 * h) * 2 + r] = acc[v];
    }
  }
  __syncthreads();
}

// ---------------------------------------------------------------------------
// cas: 2->64->32->4 tanh MLP, forward only, via WMMA (operands in LDS).
// ---------------------------------------------------------------------------
__device__ void run_cas(const float* S, float* CL1, float* CL2, float* CO,
                        const float* scW1, const float* scW2,
                        const float* scW3, const float* scb1,
                        const float* scb2, const float* scb3) {
  const int t = threadIdx.x;
  const int wid = t >> 5;
  const int lane = t & 31;
  const int h = lane >> 4, r = lane & 15;

  // L1: 32x64, K=2 padded to 4 -> 8 tiles
  {
    const int m0 = (wid & 1) * 16, n0 = (wid >> 1) * 16;
    v8f acc = {};
    acc = mm_tile<4>([&](int m, int k) { return S[(m0 + m) * S_S + k]; },
                     [&](int k, int n) { return scW1[k * CW1_S + n0 + n]; },
                     acc);
#pragma unroll
    for (int v = 0; v < 8; ++v)
      CL1[(m0 + v + 8 * h) * CL1_S + n0 + r] = htanh(acc[v] + scb1[n0 + r]);
  }
  __syncthreads();

  // L2: 32x32, K=64 -> 4 tiles (waves 0..3)
  if (wid < 4) {
    const int m0 = (wid & 1) * 16, n0 = (wid >> 1) * 16;
    v8f acc = {};
    acc = mm_tile<64>([&](int m, int k) { return CL1[(m0 + m) * CL1_S + k]; },
                      [&](int k, int n) { return scW2[k * CW2_S + n0 + n]; },
                      acc);
#pragma unroll
    for (int v = 0; v < 8; ++v)
      CL2[(m0 + v + 8 * h) * CL2_S + n0 + r] = htanh(acc[v] + scb2[n0 + r]);
  }
  __syncthreads();

  // L3: 32x4 (N padded to 16 with zeros), K=32 -> 2 tiles (waves 0,1)
  if (wid < 2) {
    const int m0 = wid * 16;
    v8f acc = {};
    acc = mm_tile<32>([&](int m, int k) { return CL2[(m0 + m) * CL2_S + k]; },
                      [&](int k, int n) { return scW3[k * CW3_S + n]; }, acc);
    if (r < 4) {
#pragma unroll
      for (int v = 0; v < 8; ++v)
        CO[(m0 + v + 8 * h) * 4 + r] = acc[v] + scb3[r];
    }
  }
  __syncthreads();
}

// ---------------------------------------------------------------------------
// Kernel 2: fold partials, leapfrog in mean space (3x gH + 2x cas, WMMA),
// emit per-batch shifts + global scale.
// ---------------------------------------------------------------------------
__global__ __launch_bounds__(256)
void k2_mlp(const float* __restrict__ pd, const float* __restrict__ dtp,
            const float* __restrict__ hW1, const float* __restrict__ hb1,
            const float* __restrict__ hW2, const float* __restrict__ hb2,
            const float* __restrict__ hW3, const float* __restrict__ hb3,
            const float* __restrict__ hW4,
            const float* __restrict__ cW1, const float* __restrict__ cb1,
            const float* __restrict__ cW2, const float* __restrict__ cb2,
            const float* __restrict__ cW3, const float* __restrict__ cb3,
            float* __restrict__ outp /* shiftQ[32], shiftP[32], scale */) {
  __shared__ float sW2[128 * W2_S];      // 67.6 KB
  __shared__ float sW3[128 * W3_S];      // 34.8 KB
  __shared__ float sW1[4 * W1_S];        // rows 2,3 zero
  __shared__ float sW1T[128 * W1T_S];    // cols 2..15 zero
  __shared__ float sW4[64];
  __shared__ float sB1[128], sB2[128], sB3[64];
  __shared__ float H1[32 * H12_S];
  __shared__ float H2[32 * H12_S];
  __shared__ float H3[32 * H3_S];
  __shared__ float S[32 * S_S];          // cols 2..7 zero
  __shared__ float G[64];
  __shared__ float SUM[64], SSQ[64];
  __shared__ float mq0[32], mp0[32], g10[32], g21[32], g30[32];
  __shared__ float aQ[32], cP[32], tmp32[32];
  __shared__ float CO_old[128], CO_new[128];

  const int t = threadIdx.x;
  const float dt = dtp[0];

  // --- stage: zero padded arrays; copy dense weights; fold partials ---
  for (int i = t; i < 4 * W1_S; i += 256) sW1[i] = 0.f;
  for (int i = t; i < 128 * W1T_S; i += 256) sW1T[i] = 0.f;
  for (int i = t; i < 128 * 128; i += 256) sW2[(i >> 7) * W2_S + (i & 127)] = hW2[i];
  for (int i = t; i < 128 * 64; i += 256) sW3[(i >> 6) * W3_S + (i & 63)] = hW3[i];
  if (t < 64) sW4[t] = hW4[t];
  if (t < 128) sB1[t] = hb1[t];
  if (t < 128) sB2[t] = hb2[t];
  if (t < 64) sB3[t] = hb3[t];
  if (t < 64) {
    float s = 0.f, q = 0.f;
    for (int c = 0; c < 32; ++c) {
      s += pd[(t * 32 + c) * 2 + 0];
      q += pd[(t * 32 + c) * 2 + 1];
    }
    SUM[t] = s; SSQ[t] = q;
  }
  __syncthreads();
  for (int i = t; i < 2 * 128; i += 256) sW1[(i >> 7) * W1_S + (i & 127)] = hW1[i];
  for (int i = t; i < 2 * 128; i += 256) {
    const int n = i >> 7, k = i & 127;
    sW1T[k * W1T_S + n] = hW1[i];
  }
  if (t < 32) {
    const float mq = SUM[2 * t] * INV_NQ;
    const float mp = SUM[2 * t + 1] * INV_NQ;
    mq0[t] = mq; mp0[t] = mp;
    S[t * S_S + 0] = mq; S[t * S_S + 1] = mp;
#pragma unroll
    for (int k = 2; k < 8; ++k) S[t * S_S + k] = 0.f;
  }
  __syncthreads();

  // g1 = gH(mq, mp)
  run_gH(S, H1, H2, H3, G, sW1, sW1T, sW2, sW3, sW4, sB1, sB2, sB3);
  if (t < 32) {
    g10[t] = G[t * 2 + 0];
    S[t * S_S + 1] = mp0[t] - 0.5f * dt * g10[t] * INV_NQ;   // mean(p_half)
  }
  __syncthreads();

  // g2 = gH(mq, mean(p_half))
  run_gH(S, H1, H2, H3, G, sW1, sW1T, sW2, sW3, sW4, sB1, sB2, sB3);
  if (t < 32) {
    g21[t] = G[t * 2 + 1];
    S[t * S_S + 0] = mq0[t] + dt * g21[t] * INV_NQ;          // mean(q_new)
  }
  __syncthreads();

  // g3 = gH(mean(q_new), mean(p_half))
  run_gH(S, H1, H2, H3, G, sW1, sW1T, sW2, sW3, sW4, sB1, sB2, sB3);

  // --- overlay cas weights into (now dead) H1, activations into H2 ---
  float* scW1 = H1;               // 4*68   = 272
  float* scW2 = H1 + 272;         // 64*36  = 2304
  float* scW3 = H1 + 2576;        // 32*20  = 640
  float* scb1 = H1 + 3216;        // 64
  float* scb2 = H1 + 3280;        // 32
  float* scb3 = H1 + 3312;        // 4   (total 3316 <= 4224)
  float* CL1  = H2;               // 32*68 = 2176
  float* CL2  = H2 + 2176;        // 32*36 = 1152 (total 3328 <= 4224)

  if (t < 32) {
    g30[t] = G[t * 2 + 0];
    aQ[t] = dt * g21[t] * INV_NQ;                      // q shift
    cP[t] = -0.5f * dt * (g10[t] + g30[t]) * INV_NQ;   // total p shift
    S[t * S_S + 0] = mq0[t]; S[t * S_S + 1] = mp0[t];  // old means for cas
  }
  for (int i = t; i < 4 * CW1_S; i += 256) scW1[i] = 0.f;
  for (int i = t; i < 32 * CW3_S; i += 256) scW3[i] = 0.f;
  __syncthreads();
  for (int i = t; i < 2 * 64; i += 256) scW1[(i >> 6) * CW1_S + (i & 63)] = cW1[i];
  for (int i = t; i < 64 * 32; i += 256) scW2[(i >> 5) * CW2_S + (i & 31)] = cW2[i];
  for (int i = t; i < 32 * 4; i += 256) scW3[(i >> 2) * CW3_S + (i & 3)] = cW3[i];
  if (t < 64) scb1[t] = cb1[t];
  if (t < 32) scb2[t] = cb2[t];
  if (t < 4) scb3[t] = cb3[t];
  __syncthreads();

  run_cas(S, CL1, CL2, CO_old, scW1, scW2, scW3, scb1, scb2, scb3);
  if (t < 32) { S[t * S_S + 0] = mq0[t] + aQ[t]; S[t * S_S + 1] = mp0[t] + cP[t]; }
  __syncthreads();
  run_cas(S, CL1, CL2, CO_new, scW1, scW2, scW3, scb1, scb2, scb3);

  // norm^2 analytically: sum((x+s)^2) = ssq + 2 s sum + N s^2
  if (t < 32) {
    const float a = aQ[t], c = cP[t];
    tmp32[t] = SSQ[2 * t] + 2.f * a * SUM[2 * t] + NQF * a * a +
               SSQ[2 * t + 1] + 2.f * c * SUM[2 * t + 1] + NQF * c * c;
  }
  __syncthreads();
  if (t == 0) {
    float n2 = 0.f;
    for (int b = 0; b < 32; ++b) n2 += tmp32[b];
    float es = 0.f;
    for (int i = 0; i < 128; ++i) es += CO_new[i] - CO_old[i];
    const float err = es * (1.0f / 128.0f);
    outp[64] = 1.0f - 0.1f * err / (sqrtf(n2) + 1e-10f);
  }
  if (t < 32) { outp[t] = aQ[t]; outp[32 + t] = cP[t]; }
}

// ---------------------------------------------------------------------------
// Kernel 3: out = (state + shift[b,half]) * scale.  Loads should hit L2
// (populated by k1); stores are non-temporal so they don't evict state.
// ---------------------------------------------------------------------------
__global__ __launch_bounds__(256)
void k3_apply(const v4f* __restrict__ st4, v4f* __restrict__ out4,
              const float* __restrict__ sp) {
  const long long gi = (long long)blockIdx.x * 256 + threadIdx.x;
  const int ch = (int)(gi >> 14);   // 16384 v4f per (b,c) channel
  const int b = ch >> 4;
  const int c = ch & 15;
  const float shift = (c >= 8) ? sp[32 + b] : sp[b];
  const float scale = sp[64];
  v4f v = st4[gi];
  v4f o;
  o.x = (v.x + shift) * scale;
  o.y = (v.y + shift) * scale;
  o.z = (v.z + shift) * scale;
  o.w = (v.w + shift) * scale;
  __builtin_nontemporal_store(o, &out4[gi]);
}

extern "C" void kernel_launch(void* const* d_in, const int* in_sizes, int n_in,
                              void* d_out, int out_size, void* d_ws,
                              size_t ws_size, hipStream_t stream) {
  (void)in_sizes; (void)n_in; (void)out_size; (void)ws_size;
  const float* state = (const float*)d_in[0];
  const float* dtp   = (const float*)d_in[1];
  const float* hW1 = (const float*)d_in[2];
  const float* hb1 = (const float*)d_in[3];
  const float* hW2 = (const float*)d_in[4];
  const float* hb2 = (const float*)d_in[5];
  const float* hW3 = (const float*)d_in[6];
  const float* hb3 = (const float*)d_in[7];
  const float* hW4 = (const float*)d_in[8];
  // hb4 (d_in[9]) drops out of the gradient — unused.
  const float* cW1 = (const float*)d_in[10];
  const float* cb1 = (const float*)d_in[11];
  const float* cW2 = (const float*)d_in[12];
  const float* cb2 = (const float*)d_in[13];
  const float* cW3 = (const float*)d_in[14];
  const float* cb3 = (const float*)d_in[15];

  float* ws = (float*)d_ws;
  float* pd = ws;          // 2048 * 2 floats of partials
  float* sp = ws + 4096;   // shiftQ[32], shiftP[32], scale

  k1_reduce<<<2048, 256, 0, stream>>>((const v4f*)state, pd);
  k2_mlp<<<1, 256, 0, stream>>>(pd, dtp, hW1, hb1, hW2, hb2, hW3, hb3, hW4,
                                cW1, cb1, cW2, cb2, cW3, cb3, sp);
  k3_apply<<<32768, 256, 0, stream>>>((const v4f*)state, (v4f*)d_out, sp);
}